// NT_Xent_batch_88081189306970
// MI455X (gfx1250) — compile-verified
//
#include <hip/hip_runtime.h>
#include <hip/hip_bf16.h>
#include <math.h>

typedef __attribute__((ext_vector_type(2))) float v2f;
typedef __attribute__((ext_vector_type(8))) float v8f;

#define B_ROWS 4096
#define DIM    256
#define INV_T  2.0f   // 1/TEMPERATURE

// ---------------------------------------------------------------------------
// Kernel 0: zero scratch (colsum + Gram accumulators) so atomics start clean.
// ---------------------------------------------------------------------------
__global__ void zero_kernel(float* __restrict__ p, int n) {
    int i = blockIdx.x * blockDim.x + threadIdx.x;
    if (i < n) p[i] = 0.0f;
}

// ---------------------------------------------------------------------------
// Kernel 1: one wave32 per row. Compute 1/max(||row||,eps), store it, and
// accumulate normalized column sums (LDS ds_add_f32, then one global atomic
// per column per block).  blockDim = 256 (8 waves = 8 rows), grid = (512, 3).
// ---------------------------------------------------------------------------
__global__ void rownorm_colsum_kernel(const float* __restrict__ z0,
                                      const float* __restrict__ z1,
                                      const float* __restrict__ z2,
                                      float* __restrict__ invnorm,
                                      float* __restrict__ colsum) {
    __shared__ float lcs[DIM];
    const int mat  = blockIdx.y;
    const float* z = (mat == 0) ? z0 : (mat == 1) ? z1 : z2;
    const int wave = threadIdx.x >> 5;
    const int lane = threadIdx.x & 31;
    const int row  = blockIdx.x * 8 + wave;

    lcs[threadIdx.x] = 0.0f;
    __syncthreads();

    const float* zr = z + (size_t)row * DIM;
    float v[8];
    float ss = 0.0f;
#pragma unroll
    for (int k = 0; k < 8; ++k) {            // coalesced: col = k*32 + lane
        v[k] = zr[k * 32 + lane];
        ss += v[k] * v[k];
    }
    // wave32 butterfly reduction
#pragma unroll
    for (int off = 16; off > 0; off >>= 1) ss += __shfl_xor(ss, off, 32);

    const float inv = 1.0f / fmaxf(sqrtf(ss), 1e-8f);
    if (lane == 0) invnorm[mat * B_ROWS + row] = inv;

#pragma unroll
    for (int k = 0; k < 8; ++k)
        atomicAdd(&lcs[k * 32 + lane], v[k] * inv);   // ds_add_f32
    __syncthreads();

    atomicAdd(&colsum[mat * DIM + threadIdx.x], lcs[threadIdx.x]);
}

// ---------------------------------------------------------------------------
// Kernel 2: G_mat = N^T N (256x256 per matrix) via V_WMMA_F32_16X16X4_F32.
// One wave owns one 16x16 output tile for one K-chunk of 512 rows:
//   3 mats * 256 tiles * 8 chunks = 6144 waves; 128 WMMA ops per wave.
// A[m,k] = N[r+k, c0+m], B[k,n] = N[r+k, c1+n]  (normalization fused via
// invnorm so the big normalized matrix is never materialized).
// ISA A-layout (16x4 f32): lanes 0-15 -> K={0,1}, lanes 16-31 -> K={2,3};
// B mirrored; C/D: VGPR g holds M=g (lanes 0-15) / M=g+8 (lanes 16-31).
// ---------------------------------------------------------------------------
__global__ void gram_wmma_kernel(const float* __restrict__ z0,
                                 const float* __restrict__ z1,
                                 const float* __restrict__ z2,
                                 const float* __restrict__ invnorm,
                                 float* __restrict__ G) {
    const int wglob = blockIdx.x * (blockDim.x >> 5) + (threadIdx.x >> 5);
    const int lane  = threadIdx.x & 31;

    const int chunk = wglob & 7;          // K-chunk (512 rows each)
    const int tile  = (wglob >> 3) & 255; // 16x16 tile id in 256x256 output
    const int mat   = wglob >> 11;        // matrix 0..2

    const float* z   = (mat == 0) ? z0 : (mat == 1) ? z1 : z2;
    const float* inv = invnorm + mat * B_ROWS;

    const int c0   = (tile >> 4) * 16;    // output row block (cols of N)
    const int c1   = (tile & 15) * 16;    // output col block (cols of N)
    const int mrow = lane & 15;           // M (for A) and N (for B/C/D)
    const int kb   = (lane >> 4) * 2;     // K sub-offset per half-wave

    v8f acc = {};
    const int r0 = chunk * 512;
#pragma unroll 4
    for (int r = r0; r < r0 + 512; r += 4) {
        const float ia = inv[r + kb];
        const float ib = inv[r + kb + 1];
        const float* za = z + (size_t)(r + kb) * DIM;
        const float* zb = z + (size_t)(r + kb + 1) * DIM;
        v2f a, b;
        a.x = za[c0 + mrow] * ia;
        a.y = zb[c0 + mrow] * ib;
        b.x = za[c1 + mrow] * ia;
        b.y = zb[c1 + mrow] * ib;
        acc = __builtin_amdgcn_wmma_f32_16x16x4_f32(
            /*neg_a=*/false, a, /*neg_b=*/false, b,
            /*c_mod=*/(short)0, acc, /*reuse_a=*/false, /*reuse_b=*/false);
    }

    // Reduce K-chunk partials into the global 256x256 accumulator.
    float* g = G + (size_t)mat * (DIM * DIM);
    const int mbase = c0 + (lane >> 4) * 8;
#pragma unroll
    for (int i = 0; i < 8; ++i)
        atomicAdd(&g[(mbase + i) * DIM + c1 + mrow], acc[i]);
}

// ---------------------------------------------------------------------------
// Kernel 3: loss = inv_t^2 * sum(G0^2+G1^2+G2^2 + G0G1+G0G2+G1G2)
//                - 2*inv_t * sum(colsum^2) + 3*B^2
// ---------------------------------------------------------------------------
__global__ void finalize_kernel(const float* __restrict__ G,
                                const float* __restrict__ colsum,
                                float* __restrict__ out) {
    const int tid = threadIdx.x;
    const float* g0 = G;
    const float* g1 = G + DIM * DIM;
    const float* g2 = G + 2 * DIM * DIM;

    float s1 = 0.0f, s2 = 0.0f;
    for (int i = tid; i < DIM * DIM; i += 256) {
        const float a = g0[i], b = g1[i], c = g2[i];
        s1 += a * a + b * b + c * c + a * b + a * c + b * c;
    }
    for (int d = tid; d < 3 * DIM; d += 256) {
        const float cs = colsum[d];
        s2 += cs * cs;
    }
#pragma unroll
    for (int off = 16; off > 0; off >>= 1) {
        s1 += __shfl_xor(s1, off, 32);
        s2 += __shfl_xor(s2, off, 32);
    }
    __shared__ float r1[8], r2[8];
    if ((tid & 31) == 0) { r1[tid >> 5] = s1; r2[tid >> 5] = s2; }
    __syncthreads();
    if (tid == 0) {
        float S1 = 0.0f, S2 = 0.0f;
#pragma unroll
        for (int w = 0; w < 8; ++w) { S1 += r1[w]; S2 += r2[w]; }
        const float posConst = 3.0f * (float)B_ROWS * (float)B_ROWS;
        out[0] = (INV_T * INV_T) * S1 - 2.0f * INV_T * S2 + posConst;
    }
}

// ---------------------------------------------------------------------------
extern "C" void kernel_launch(void* const* d_in, const int* in_sizes, int n_in,
                              void* d_out, int out_size, void* d_ws, size_t ws_size,
                              hipStream_t stream) {
    const float* z0 = (const float*)d_in[0];
    const float* z1 = (const float*)d_in[1];
    const float* z2 = (const float*)d_in[2];

    float* ws      = (float*)d_ws;
    float* invnorm = ws;                           // 3*4096 floats
    float* colsum  = invnorm + 3 * B_ROWS;         // 3*256 floats
    float* G       = colsum + 3 * DIM;             // 3*65536 floats
    // total scratch: ~840 KB

    const int nzero = 3 * DIM + 3 * DIM * DIM;     // colsum + G (contiguous)
    zero_kernel<<<(nzero + 255) / 256, 256, 0, stream>>>(colsum, nzero);

    dim3 gridN(B_ROWS / 8, 3);
    rownorm_colsum_kernel<<<gridN, 256, 0, stream>>>(z0, z1, z2, invnorm, colsum);

    // 6144 waves, 8 waves per 256-thread block
    gram_wmma_kernel<<<6144 / 8, 256, 0, stream>>>(z0, z1, z2, invnorm, G);

    finalize_kernel<<<1, 256, 0, stream>>>(G, colsum, (float*)d_out);
}